// KeypointRenderer_37804302139574
// MI455X (gfx1250) — compile-verified
//
#include <hip/hip_runtime.h>

// KeypointRenderer heatmap: out[n,0,y,x] = max_k gy[n,k,y]*gx[n,k,x]
//                                       = exp(-min_k (sy[k][y] + sx[k][x]))
// sy/sx are masked scaled squared distances (invalid -> BIG -> exp underflows to 0,
// matching the reference's masked-zero semantics).
//
// The 16x16 broadcast-sum sy_i + sx_j is computed on the matrix pipe as a
// rank-2 f32 WMMA: A = [sy | 1 | 0 | 0] (16x4), B = [1 ; sx ; 0 ; 0] (4x16).

#define HH 96
#define WW 96
#define KK 21
#define BIGV 1.0e30f
#define INV2S2 (1.0f / 18.0f)

typedef __attribute__((ext_vector_type(2))) float v2f;
typedef __attribute__((ext_vector_type(8))) float v8f;

typedef __attribute__((address_space(1))) int* gptr_i32;
typedef __attribute__((address_space(3))) int* lptr_i32;

#if __has_builtin(__builtin_amdgcn_global_load_async_to_lds_b32)
#define HAVE_ASYNC_LDS 1
#else
#define HAVE_ASYNC_LDS 0
#endif

__global__ __launch_bounds__(256) void kp_heat_wmma_kernel(
    const float* __restrict__ keypoints,   // [N][K][3]
    const float* __restrict__ boxes,       // [N][4]
    float* __restrict__ out)               // [N][1][96][96]
{
    __shared__ float s_sx[KK * WW];   // masked c*dx^2 (includes vis gate)
    __shared__ float s_sy[KK * HH];   // masked c*dy^2
    __shared__ float s_kp[KK * 3 + 4];

    const int n   = blockIdx.x;
    const int tid = threadIdx.x;
    const float* kp = keypoints + (size_t)n * (KK * 3);
    const float* hb = boxes + (size_t)n * 4;

    // ---- Stage keypoints + box into LDS (CDNA5 async-to-LDS path if present) ----
#if HAVE_ASYNC_LDS
    if (tid < KK * 3 + 4) {
        const float* src = (tid < KK * 3) ? (kp + tid) : (hb + (tid - KK * 3));
        __builtin_amdgcn_global_load_async_to_lds_b32(
            (gptr_i32)src,
            (lptr_i32)&s_kp[tid],
            0, 0);
    }
#if __has_builtin(__builtin_amdgcn_s_wait_asynccnt)
    __builtin_amdgcn_s_wait_asynccnt(0);
#else
    asm volatile("s_wait_asynccnt 0" ::: "memory");
#endif
#else
    if (tid < KK * 3 + 4) {
        s_kp[tid] = (tid < KK * 3) ? kp[tid] : hb[tid - KK * 3];
    }
#endif
    __syncthreads();

    const float x0 = s_kp[KK * 3 + 0];
    const float y0 = s_kp[KK * 3 + 1];
    const float bw = s_kp[KK * 3 + 2] - x0;
    const float bh = s_kp[KK * 3 + 3] - y0;

    // ---- Phase 1: build sx[k][x], sy[k][y] tables ----
    for (int e = tid; e < 2 * KK * 96; e += 256) {
        const bool isY = (e >= KK * 96);
        const int  r   = isY ? (e - KK * 96) : e;
        const int  k   = r / 96;
        const int  p   = r - k * 96;

        const float kx = s_kp[k * 3 + 0];
        const float ky = s_kp[k * 3 + 1];
        const float kv = s_kp[k * 3 + 2];

        const float org = isY ? y0 : x0;
        const float ext = isY ? bh : bw;
        const float kc  = isY ? ky : kx;
        const float c   = (kc - org) / ext * 96.0f;    // xs or ys

        const float ul = truncf(c - 9.0f);             // trunc(c - 3*sigma)
        const float br = truncf(c + 10.0f);            // trunc(c + 3*sigma + 1)
        const float pf = (float)p;
        const float d  = pf - ul - 9.0f;

        bool m = (pf >= fmaxf(ul, 0.0f)) && (pf < fminf(br, 96.0f));
        if (!isY) m = m && (kv > 0.1f);                // vis gates gx only

        const float s = m ? (d * d * INV2S2) : BIGV;
        if (isY) s_sy[r] = s; else s_sx[r] = s;
    }
    __syncthreads();

    // ---- Phase 2: per-tile min_k (sy_i + sx_j) via V_WMMA_F32_16X16X4_F32 ----
    const int  lane  = tid & 31;
    const int  wave  = tid >> 5;
    const bool lolan = (lane < 16);
    const int  col   = lane & 15;
    const float one_lo = lolan ? 1.0f : 0.0f;

    for (int t = wave; t < 36; t += 8) {               // 6x6 tiles of 16x16
        const int ty = t / 6;
        const int tx = t - ty * 6;
        const int rb = ty * 16;
        const int cb = tx * 16;

        float mn[8];
#pragma unroll
        for (int r = 0; r < 8; ++r) mn[r] = 4.0f * BIGV;

        for (int k = 0; k < KK; ++k) {
            const float syv = s_sy[k * 96 + rb + col]; // A row M = lane&15
            const float sxv = s_sx[k * 96 + cb + col]; // B col N = lane&15

            v2f a, b;
            // A 16x4: lanes 0-15 hold K=0 (v0) and K=1 (v1); lanes 16-31 hold K=2,3 (zeros)
            a.x = lolan ? syv : 0.0f;
            a.y = one_lo;
            // B 4x16: lanes 0-15 hold K=0 row (v0) and K=1 row (v1); lanes 16-31 hold K=2,3
            b.x = one_lo;
            b.y = lolan ? sxv : 0.0f;

            v8f cz = {0.0f, 0.0f, 0.0f, 0.0f, 0.0f, 0.0f, 0.0f, 0.0f};
            v8f d = __builtin_amdgcn_wmma_f32_16x16x4_f32(
                false, a, false, b, (short)0, cz, false, false);

#pragma unroll
            for (int r = 0; r < 8; ++r) mn[r] = fminf(mn[r], d[r]);
        }

        // D layout: lanes 0-15 -> rows rb+0..7, lanes 16-31 -> rows rb+8..15; col = cb + (lane&15)
        const int yb = rb + (lolan ? 0 : 8);
        float* po = out + (size_t)n * (HH * WW) + (size_t)yb * WW + cb + col;
#pragma unroll
        for (int r = 0; r < 8; ++r) {
            po[r * WW] = __expf(-mn[r]);
        }
    }
}

extern "C" void kernel_launch(void* const* d_in, const int* in_sizes, int n_in,
                              void* d_out, int out_size, void* d_ws, size_t ws_size,
                              hipStream_t stream) {
    const float* keypoints = (const float*)d_in[0];   // [N][21][3]
    const float* boxes     = (const float*)d_in[1];   // [N][4]
    float* out             = (float*)d_out;           // [N][1][96][96]
    const int N = in_sizes[1] / 4;                    // 1024
    kp_heat_wmma_kernel<<<dim3(N), dim3(256), 0, stream>>>(keypoints, boxes, out);
}